// PatientMeanEncoder_16612933501067
// MI455X (gfx1250) — compile-verified
//
#include <hip/hip_runtime.h>
#include <hip/hip_bf16.h>

typedef __attribute__((ext_vector_type(2))) float v2f;
typedef __attribute__((ext_vector_type(8))) float v8f;

#define NPAT 64
#define LSTEP 2048
#define CIN 256
#define COUT 276
#define SEG 128
#define NSEG 16   // SEG*NSEG == LSTEP

// ---------------------------------------------------------------------------
// Kernel A: dem MLP via fp32 WMMA.  d = relu(relu(dem@w1+b1)@w2+b2)  -> [64,20]
// One workgroup, 4 waves; wave w owns M-tile rows [16w, 16w+16).
// ---------------------------------------------------------------------------
__global__ void __launch_bounds__(128)
mlp_wmma_kernel(const float* __restrict__ dem, const float* __restrict__ w1,
                const float* __restrict__ b1, const float* __restrict__ w2,
                const float* __restrict__ b2, float* __restrict__ d20) {
  __shared__ float h[NPAT][40];          // layer-1 activations
  const int wave = threadIdx.x >> 5;     // 0..3  -> M tile
  const int lane = threadIdx.x & 31;
  const int half = lane >> 4;            // 0: lanes 0-15, 1: lanes 16-31
  const int lrow = lane & 15;
  const int mbase = wave * 16;

  // ---- Layer 1: [64,10] @ [10,40] + b1, K padded 10->12, N padded 40->48
  for (int nt = 0; nt < 3; ++nt) {
    const int col = nt * 16 + lrow;                  // N index (B/C/D lane dim)
    const float bias = (col < 40) ? b1[col] : 0.0f;
    v8f acc;
    #pragma unroll
    for (int i = 0; i < 8; ++i) acc[i] = bias;       // C(m,n) = b1[n]
    for (int kt = 0; kt < 3; ++kt) {
      const int am = mbase + lrow;                   // A: lane dim = M
      const int k0 = kt * 4 + (half ? 2 : 0);
      const int k1 = k0 + 1;
      v2f a, b;
      a[0] = (k0 < 10) ? dem[am * 10 + k0] : 0.0f;
      a[1] = (k1 < 10) ? dem[am * 10 + k1] : 0.0f;
      b[0] = (col < 40 && k0 < 10) ? w1[k0 * 40 + col] : 0.0f;
      b[1] = (col < 40 && k1 < 10) ? w1[k1 * 40 + col] : 0.0f;
      acc = __builtin_amdgcn_wmma_f32_16x16x4_f32(false, a, false, b,
                                                  (short)0, acc, false, false);
    }
    if (col < 40) {
      #pragma unroll
      for (int r = 0; r < 8; ++r)
        h[mbase + half * 8 + r][col] = fmaxf(acc[r], 0.0f);
    }
  }
  __syncthreads();

  // ---- Layer 2: [64,40] @ [40,20] + b2, K=40 exact (10 k-tiles), N padded 20->32
  for (int nt = 0; nt < 2; ++nt) {
    const int col = nt * 16 + lrow;
    const float bias = (col < 20) ? b2[col] : 0.0f;
    v8f acc;
    #pragma unroll
    for (int i = 0; i < 8; ++i) acc[i] = bias;
    for (int kt = 0; kt < 10; ++kt) {
      const int am = mbase + lrow;
      const int k0 = kt * 4 + (half ? 2 : 0);
      const int k1 = k0 + 1;
      v2f a, b;
      a[0] = h[am][k0];
      a[1] = h[am][k1];
      b[0] = (col < 20) ? w2[k0 * 20 + col] : 0.0f;
      b[1] = (col < 20) ? w2[k1 * 20 + col] : 0.0f;
      acc = __builtin_amdgcn_wmma_f32_16x16x4_f32(false, a, false, b,
                                                  (short)0, acc, false, false);
    }
    if (col < 20) {
      #pragma unroll
      for (int r = 0; r < 8; ++r)
        d20[(mbase + half * 8 + r) * 20 + col] = fmaxf(acc[r], 0.0f);
    }
  }
}

// ---------------------------------------------------------------------------
// Kernel B: per-segment sum / nonzero-count partials over timesteps.
// 262144 threads: tid -> (seg, n, c); consecutive tids span c (coalesced).
// ---------------------------------------------------------------------------
__global__ void __launch_bounds__(256)
seg_partials_kernel(const float* __restrict__ ts, float* __restrict__ psum,
                    float* __restrict__ pcnt) {
  const int tid = blockIdx.x * blockDim.x + threadIdx.x;
  const int chains = NPAT * CIN;
  const int s = tid / chains;
  const int chain = tid % chains;
  const int n = chain / CIN;
  const int c = chain % CIN;
  if (s >= NSEG) return;

  const float* p = ts + ((size_t)n * LSTEP + (size_t)s * SEG) * CIN + c;
  float sum = 0.0f, cnt = 0.0f;
  for (int i = 0; i < SEG; ++i) {
    if (i + 16 < SEG) __builtin_prefetch(p + (size_t)(i + 16) * CIN, 0, 0);
    const float v = p[(size_t)i * CIN];
    sum += v;
    cnt += (v != 0.0f) ? 1.0f : 0.0f;
  }
  psum[(size_t)chain * NSEG + s] = sum;
  pcnt[(size_t)chain * NSEG + s] = cnt;
}

// ---------------------------------------------------------------------------
// Kernel C: re-walk each segment with the exclusive-prefix offset and emit
// relu(prefix_sum / max(prefix_count,1)) for channels 0..255.
// Second read of ts hits the 192 MB L2 (input is 134 MB).
// ---------------------------------------------------------------------------
__global__ void __launch_bounds__(256)
scan_write_kernel(const float* __restrict__ ts, const float* __restrict__ psum,
                  const float* __restrict__ pcnt, float* __restrict__ out) {
  const int tid = blockIdx.x * blockDim.x + threadIdx.x;
  const int chains = NPAT * CIN;
  const int s = tid / chains;
  const int chain = tid % chains;
  const int n = chain / CIN;
  const int c = chain % CIN;
  if (s >= NSEG) return;

  float sum = 0.0f, cnt = 0.0f;
  const float* ps = psum + (size_t)chain * NSEG;
  const float* pc = pcnt + (size_t)chain * NSEG;
  for (int s2 = 0; s2 < s; ++s2) { sum += ps[s2]; cnt += pc[s2]; }

  const float* p = ts + ((size_t)n * LSTEP + (size_t)s * SEG) * CIN + c;
  float* o = out + ((size_t)n * LSTEP + (size_t)s * SEG) * COUT + c;
  for (int i = 0; i < SEG; ++i) {
    const float v = p[(size_t)i * CIN];
    sum += v;
    cnt += (v != 0.0f) ? 1.0f : 0.0f;
    const float avg = sum / fmaxf(cnt, 1.0f);
    o[(size_t)i * COUT] = fmaxf(avg, 0.0f);
  }
}

// ---------------------------------------------------------------------------
// Kernel D: dem channels are constant over L -> prefix mean is the value
// itself (zeros stay zero), and d>=0 so the final relu is identity.
// Pure broadcast: out[n, l, 256+j] = d[n, j].
// ---------------------------------------------------------------------------
__global__ void __launch_bounds__(256)
bcast_dem_kernel(const float* __restrict__ d20, float* __restrict__ out) {
  const int tid = blockIdx.x * blockDim.x + threadIdx.x;
  const int total = NPAT * LSTEP * 20;
  if (tid >= total) return;
  const int j = tid % 20;
  const int rem = tid / 20;
  const int l = rem % LSTEP;
  const int n = rem / LSTEP;
  out[((size_t)n * LSTEP + l) * COUT + CIN + j] = d20[n * 20 + j];
}

// ---------------------------------------------------------------------------
extern "C" void kernel_launch(void* const* d_in, const int* in_sizes, int n_in,
                              void* d_out, int out_size, void* d_ws, size_t ws_size,
                              hipStream_t stream) {
  const float* dem = (const float*)d_in[0];   // [64,10]
  const float* ts  = (const float*)d_in[1];   // [64,2048,256]
  const float* w1  = (const float*)d_in[2];   // [10,40]
  const float* b1  = (const float*)d_in[3];   // [40]
  const float* w2  = (const float*)d_in[4];   // [40,20]
  const float* b2  = (const float*)d_in[5];   // [20]
  float* out = (float*)d_out;                 // [64,2048,276]

  // Workspace layout: d20 (5 KB) | psum (1 MB) | pcnt (1 MB)
  float* d20  = (float*)d_ws;
  float* psum = (float*)((char*)d_ws + 8192);
  float* pcnt = (float*)((char*)d_ws + 8192 + (size_t)NPAT * CIN * NSEG * 4);

  mlp_wmma_kernel<<<1, 128, 0, stream>>>(dem, w1, b1, w2, b2, d20);

  const int scan_threads = NPAT * CIN * NSEG;          // 262144
  seg_partials_kernel<<<scan_threads / 256, 256, 0, stream>>>(ts, psum, pcnt);
  scan_write_kernel<<<scan_threads / 256, 256, 0, stream>>>(ts, psum, pcnt, out);

  const int bc_threads = NPAT * LSTEP * 20;            // 2621440
  bcast_dem_kernel<<<(bc_threads + 255) / 256, 256, 0, stream>>>(d20, out);
}